// FeatureAlign_R2_new3_89180700934263
// MI455X (gfx1250) — compile-verified
//
#include <hip/hip_runtime.h>
#include <hip/hip_bf16.h>

typedef float v2f __attribute__((ext_vector_type(2)));
typedef float v8f __attribute__((ext_vector_type(8)));

__device__ __forceinline__ v8f wmma_f32(v2f a, v2f b, v8f c) {
    // D = A(16x4, f32) x B(4x16, f32) + C(16x16, f32)
    return __builtin_amdgcn_wmma_f32_16x16x4_f32(false, a, false, b, (short)0, c,
                                                 false, false);
}

// ---------------------------------------------------------------------------
// Bilinear resize (half-pixel centers, matches align_corners=False upsampling)
// ---------------------------------------------------------------------------
__global__ void k_bilinear(const float* __restrict__ in, float* __restrict__ out,
                           int B, int C, int Hin, int Win, int Hout, int Wout,
                           int inBS, int outBS) {
    int i = blockIdx.x * blockDim.x + threadIdx.x;
    int total = B * C * Hout * Wout;
    if (i >= total) return;
    int ox = i % Wout; int t = i / Wout;
    int oy = t % Hout; t /= Hout;
    int c  = t % C;    int b = t / C;
    float sy = (oy + 0.5f) * ((float)Hin / (float)Hout) - 0.5f;
    float sx = (ox + 0.5f) * ((float)Win / (float)Wout) - 0.5f;
    float y0f = floorf(sy), x0f = floorf(sx);
    float fy = sy - y0f,    fx = sx - x0f;
    int y0 = (int)y0f, x0 = (int)x0f;
    int y0c = min(max(y0, 0), Hin - 1), y1c = min(max(y0 + 1, 0), Hin - 1);
    int x0c = min(max(x0, 0), Win - 1), x1c = min(max(x0 + 1, 0), Win - 1);
    const float* p = in + (long)b * inBS + (long)c * Hin * Win;
    float v00 = p[y0c * Win + x0c], v01 = p[y0c * Win + x1c];
    float v10 = p[y1c * Win + x0c], v11 = p[y1c * Win + x1c];
    float v = (1.f - fy) * ((1.f - fx) * v00 + fx * v01)
            +        fy  * ((1.f - fx) * v10 + fx * v11);
    out[(long)b * outBS + (long)c * Hout * Wout + oy * Wout + ox] = v;
}

// ---------------------------------------------------------------------------
// 1x1 conv as WMMA GEMM: out[b, co, p] = sum_ci W[co,ci] * X[ci, p]
//   ci in [0,256) -> inA, ci in [256,512) -> inB (scaleB folded into A)
// HW is compile-time -> channel strides become immediate load offsets.
// Wave tile = 16(co) x 32(pix): shared A fragment, two accumulators.
// When !HAS_SA the B loads feed WMMA directly (no VALU between load and MMA).
// ---------------------------------------------------------------------------
template <int HW, bool HAS_SA>
__global__ void k_conv1x1_wmma(const float* __restrict__ W,
                               const float* __restrict__ inA,
                               const float* __restrict__ inB,
                               int lda, int ldb, float scaleB,
                               const float* __restrict__ sa,
                               float* __restrict__ out, int outBS, int B) {
    int wave = threadIdx.x >> 5, lane = threadIdx.x & 31;
    int tid  = blockIdx.x * (blockDim.x >> 5) + wave;
    const int hwT = HW >> 5;
    const int perB = 16 * hwT;
    if (tid >= B * perB) return;
    int b   = tid / perB, r = tid % perB;
    int co0 = (r / hwT) << 4;
    int p0  = (r % hwT) << 5;
    int m   = lane & 15;
    int kb  = (lane >> 4) << 1;
    const float* Arow = W + (long)(co0 + m) * 512;
    const float* pA   = inA + (long)b * lda + p0 + m;
    const float* pB   = inB + (long)b * ldb + p0 + m;
    float ss0 = 1.0f, ss1 = 1.0f;
    if (HAS_SA) {
        ss0 = 1.0f + sa[(long)b * HW + p0 + m];
        ss1 = 1.0f + sa[(long)b * HW + p0 + 16 + m];
    }
    v8f acc0 = {}, acc1 = {};
#pragma unroll 4
    for (int k0 = 0; k0 < 256; k0 += 4) {
        __builtin_prefetch(Arow + k0 + 64, 0, 0);
        int ci = k0 + kb;
        v2f a, b0, b1;
        a.x  = Arow[k0 + kb];
        a.y  = Arow[k0 + kb + 1];
        b0.x = pA[ci * HW];
        b0.y = pA[ci * HW + HW];
        b1.x = pA[ci * HW + 16];
        b1.y = pA[ci * HW + HW + 16];
        if (HAS_SA) { b0.x *= ss0; b0.y *= ss0; b1.x *= ss1; b1.y *= ss1; }
        acc0 = wmma_f32(a, b0, acc0);
        acc1 = wmma_f32(a, b1, acc1);
    }
#pragma unroll 4
    for (int k0 = 0; k0 < 256; k0 += 4) {
        __builtin_prefetch(Arow + 256 + k0 + 64, 0, 0);
        int ci = k0 + kb;
        v2f a, b0, b1;
        a.x  = Arow[256 + k0 + kb] * scaleB;   // scaleB folded into A
        a.y  = Arow[256 + k0 + kb + 1] * scaleB;
        b0.x = pB[ci * HW];
        b0.y = pB[ci * HW + HW];
        b1.x = pB[ci * HW + 16];
        b1.y = pB[ci * HW + HW + 16];
        if (HAS_SA) { b0.x *= ss0; b0.y *= ss0; b1.x *= ss1; b1.y *= ss1; }
        acc0 = wmma_f32(a, b0, acc0);
        acc1 = wmma_f32(a, b1, acc1);
    }
    int rowOff = (lane >> 4) << 3;
    float* o = out + (long)b * outBS + p0 + m;
#pragma unroll
    for (int r8 = 0; r8 < 8; ++r8) {
        long row = (long)(co0 + rowOff + r8) * HW;
        o[row]      = acc0[r8];
        o[row + 16] = acc1[r8];
    }
}

// ---------------------------------------------------------------------------
// 3x3 conv (pad 1) 256 -> 216 (offset/mask conv), im2col-style WMMA GEMM.
// Branch-free: OOB pixels / dead co rows via clamped addresses + zero flags.
// Wave tile = 16(co, padded to 224) x 32(pix). H, W compile-time.
// ---------------------------------------------------------------------------
template <int H, int Wd>
__global__ void k_conv3x3_wmma(const float* __restrict__ Wm,   // [216,256,3,3]
                               const float* __restrict__ bias, // [216]
                               const float* __restrict__ x,    // [B,256,H,W]
                               float* __restrict__ out,        // [B,216,H,W]
                               int B) {
    const int HW = H * Wd;
    int wave = threadIdx.x >> 5, lane = threadIdx.x & 31;
    int tid  = blockIdx.x * (blockDim.x >> 5) + wave;
    const int hwT = HW >> 5;
    const int perB = 14 * hwT;
    if (tid >= B * perB) return;
    int b   = tid / perB, r = tid % perB;
    int co0 = (r / hwT) << 4;
    int p0  = (r % hwT) << 5;
    int m   = lane & 15;
    int kb  = (lane >> 4) << 1;
    int y   = p0 / Wd, x0p = p0 % Wd;
    int xn0 = x0p + m, xn1 = x0p + 16 + m;
    int coA = co0 + m;
    float af = (coA < 216) ? 1.f : 0.f;
    const float* wbase = Wm + (long)min(coA, 215) * 256 * 9;
    const float* xb = x + (long)b * 256 * HW;
    v8f acc0 = {}, acc1 = {};
    for (int t = 0; t < 9; ++t) {
        int ky = t / 3 - 1, kx = t % 3 - 1;
        int ys = y + ky;
        bool yv = (ys >= 0) && (ys < H);
        int ysc = min(max(ys, 0), H - 1);
        int xs0 = xn0 + kx, xs1 = xn1 + kx;
        float f0 = (yv && xs0 >= 0 && xs0 < Wd) ? af * 0.f + 1.f : 0.f;
        float f1 = (yv && xs1 >= 0 && xs1 < Wd) ? 1.f : 0.f;
        f0 = (yv && xs0 >= 0 && xs0 < Wd) ? 1.f : 0.f;
        int xs0c = min(max(xs0, 0), Wd - 1);
        int xs1c = min(max(xs1, 0), Wd - 1);
        const float* x0r = xb + ysc * Wd + xs0c;
        const float* x1r = xb + ysc * Wd + xs1c;
        const float* wr  = wbase + t;
#pragma unroll 4
        for (int c0 = 0; c0 < 256; c0 += 4) {
            int ci = c0 + kb;
            v2f a, b0, b1;
            a.x  = wr[ci * 9] * af;
            a.y  = wr[ci * 9 + 9] * af;
            b0.x = x0r[ci * HW] * f0;
            b0.y = x0r[ci * HW + HW] * f0;
            b1.x = x1r[ci * HW] * f1;
            b1.y = x1r[ci * HW + HW] * f1;
            acc0 = wmma_f32(a, b0, acc0);
            acc1 = wmma_f32(a, b1, acc1);
        }
    }
    int rowOff = (lane >> 4) << 3;
    float* o = out + (long)b * 216 * HW + p0 + m;
#pragma unroll
    for (int r8 = 0; r8 < 8; ++r8) {
        int row = co0 + rowOff + r8;
        if (row < 216) {
            float bz = bias[row];
            o[(long)row * HW]      = acc0[r8] + bz;
            o[(long)row * HW + 16] = acc1[r8] + bz;
        }
    }
}

// ---------------------------------------------------------------------------
// Modulated deformable conv v2 (3x3, pad 1, dg=8) + bias + ReLU.
// Block = 256 threads (8 waves) computes 128 co x 32 pixels.
// Per (group g, tap k): gather modulated bilinear val[32ch x 32pix] into LDS,
// then each wave runs 16 f32 WMMAs (K=32, two N-halves sharing A fragment).
// ---------------------------------------------------------------------------
template <int H, int Wi>
__global__ void k_dcn_wmma(const float* __restrict__ x,    // [B,256,H,W] feat_up
                           const float* __restrict__ om,   // [B,216,H,W]
                           const float* __restrict__ Wd,   // [256,256,3,3]
                           const float* __restrict__ bias, // [256]
                           float* __restrict__ out,        // [B,256,H,W]
                           int B) {
    __shared__ float vtile[32 * 32];      // [c][pix]
    const int HW  = H * Wi;
    const int hwT = HW >> 5;
    int blk = blockIdx.x;                 // B * 2 * hwT blocks
    int p0  = (blk % hwT) << 5;  blk /= hwT;
    int half = blk & 1;  int b = blk >> 1;
    int t    = threadIdx.x;
    int wave = t >> 5, lane = t & 31;
    int m    = lane & 15, kb = (lane >> 4) << 1;
    int co0  = half * 128 + wave * 16;
    int pix  = t & 31;
    int c8   = t >> 5;                    // stages channels c8, c8+8, c8+16, c8+24
    int p    = p0 + pix;
    int py   = p / Wi, px = p % Wi;
    const float* xb  = x  + (long)b * 256 * HW;
    const float* omb = om + (long)b * 216 * HW + p;
    const float* wb  = Wd + (long)(co0 + m) * 256 * 9;
    v8f acc0 = {}, acc1 = {};
    for (int g = 0; g < 8; ++g) {
        for (int k = 0; k < 9; ++k) {
            float oy  = omb[(g * 9 + k) * HW];
            float ox  = omb[(72 + g * 9 + k) * HW];
            float mr  = omb[(144 + g * 9 + k) * HW];
            float msk = 1.0f / (1.0f + __expf(-mr));
            float syf = (float)py + (float)(k / 3 - 1) + oy;
            float sxf = (float)px + (float)(k % 3 - 1) + ox;
            float y0f = floorf(syf), x0f = floorf(sxf);
            float fy = syf - y0f, fx = sxf - x0f;
            int y0 = (int)y0f, x0i = (int)x0f;
            int y1 = y0 + 1,  x1  = x0i + 1;
            float vy0 = (y0 >= 0 && y0 < H)    ? 1.f : 0.f;
            float vy1 = (y1 >= 0 && y1 < H)    ? 1.f : 0.f;
            float vx0 = (x0i >= 0 && x0i < Wi) ? 1.f : 0.f;
            float vx1 = (x1 >= 0 && x1 < Wi)   ? 1.f : 0.f;
            int y0c = min(max(y0, 0), H - 1),   y1c = min(max(y1, 0), H - 1);
            int x0c = min(max(x0i, 0), Wi - 1), x1c = min(max(x1, 0), Wi - 1);
            float w00 = (1.f - fy) * (1.f - fx) * vy0 * vx0;
            float w01 = (1.f - fy) * fx * vy0 * vx1;
            float w10 = fy * (1.f - fx) * vy1 * vx0;
            float w11 = fy * fx * vy1 * vx1;
            int i00 = y0c * Wi + x0c, i01 = y0c * Wi + x1c;
            int i10 = y1c * Wi + x0c, i11 = y1c * Wi + x1c;
            __syncthreads();   // protect vtile from previous iteration's readers
#pragma unroll
            for (int cc = 0; cc < 4; ++cc) {
                int c = c8 + cc * 8;
                const float* xc = xb + (g * 32 + c) * HW;
                float v = w00 * xc[i00] + w01 * xc[i01]
                        + w10 * xc[i10] + w11 * xc[i11];
                vtile[c * 32 + pix] = msk * v;
            }
            __syncthreads();
            const float* wr = wb + (g * 32 + kb) * 9 + k;
#pragma unroll
            for (int kk = 0; kk < 32; kk += 4) {
                v2f a, b0, b1;
                a.x  = wr[kk * 9];
                a.y  = wr[kk * 9 + 9];
                b0.x = vtile[(kk + kb) * 32 + m];
                b0.y = vtile[(kk + kb + 1) * 32 + m];
                b1.x = vtile[(kk + kb) * 32 + 16 + m];
                b1.y = vtile[(kk + kb + 1) * 32 + 16 + m];
                acc0 = wmma_f32(a, b0, acc0);
                acc1 = wmma_f32(a, b1, acc1);
            }
        }
    }
    int rowOff = (lane >> 4) << 3;
    float* o = out + (long)b * 256 * HW + p0 + m;
#pragma unroll
    for (int r8 = 0; r8 < 8; ++r8) {
        int row = co0 + rowOff + r8;
        float bz = bias[row];
        o[(long)row * HW]      = fmaxf(acc0[r8] + bz, 0.f);
        o[(long)row * HW + 16] = fmaxf(acc1[r8] + bz, 0.f);
    }
}

// ---------------------------------------------------------------------------
// Channel-wise mean & max over C=512 of feat2 -> avg[B*HW], max[B*HW]
// ---------------------------------------------------------------------------
__global__ void k_avgmax(const float* __restrict__ feat2, float* __restrict__ avg,
                         float* __restrict__ mx, int B, int HW) {
    int i = blockIdx.x * blockDim.x + threadIdx.x;
    if (i >= B * HW) return;
    int b = i / HW, p = i % HW;
    const float* base = feat2 + (long)b * 512 * HW + p;
    float s = 0.f, m = -3.4e38f;
    for (int c = 0; c < 512; ++c) {
        float v = base[(long)c * HW];
        s += v;
        m = fmaxf(m, v);
    }
    avg[i] = s * (1.0f / 512.0f);
    mx[i]  = m;
}

// ---------------------------------------------------------------------------
// 7x7 conv over [avg;max] (pad 3) + BatchNorm + sigmoid -> sa[B*HW]
// ---------------------------------------------------------------------------
__global__ void k_sa(const float* __restrict__ avg, const float* __restrict__ mx,
                     const float* __restrict__ w_sa,
                     const float* __restrict__ bn_g, const float* __restrict__ bn_b,
                     const float* __restrict__ bn_m, const float* __restrict__ bn_v,
                     float* __restrict__ sa, int B, int H, int W) {
    int i = blockIdx.x * blockDim.x + threadIdx.x;
    if (i >= B * H * W) return;
    int p = i % (H * W), b = i / (H * W);
    int y = p / W, x = p % W;
    float acc = 0.f;
    for (int c = 0; c < 2; ++c) {
        const float* inp = (c == 0 ? avg : mx) + (long)b * H * W;
        for (int ky = 0; ky < 7; ++ky) {
            int ys = y + ky - 3;
            if (ys < 0 || ys >= H) continue;
            for (int kx = 0; kx < 7; ++kx) {
                int xs = x + kx - 3;
                if (xs < 0 || xs >= W) continue;
                acc += w_sa[(c * 7 + ky) * 7 + kx] * inp[ys * W + xs];
            }
        }
    }
    float a = (acc - bn_m[0]) * rsqrtf(bn_v[0] + 1e-5f);
    a = a * bn_g[0] + bn_b[0];
    sa[i] = 1.0f / (1.0f + __expf(-a));
}

// ---------------------------------------------------------------------------
// GroupNorm (32 groups): per-(b,g) mean & rsqrt(var) via block reduction
// ---------------------------------------------------------------------------
__global__ void k_gn_reduce(const float* __restrict__ y, float* __restrict__ gn,
                            int HW) {
    int bg = blockIdx.x;                  // B*32 blocks
    int b = bg >> 5, g = bg & 31;
    const float* base = y + (long)b * 256 * HW + (long)g * 8 * HW;
    int n = 8 * HW;
    float s = 0.f, ss = 0.f;
    for (int i = threadIdx.x; i < n; i += blockDim.x) {
        float v = base[i];
        s += v;
        ss += v * v;
    }
    __shared__ float sh0[256], sh1[256];
    sh0[threadIdx.x] = s;  sh1[threadIdx.x] = ss;
    __syncthreads();
    for (int st = 128; st > 0; st >>= 1) {
        if (threadIdx.x < st) {
            sh0[threadIdx.x] += sh0[threadIdx.x + st];
            sh1[threadIdx.x] += sh1[threadIdx.x + st];
        }
        __syncthreads();
    }
    if (threadIdx.x == 0) {
        float mu  = sh0[0] / (float)n;
        float var = sh1[0] / (float)n - mu * mu;
        gn[bg * 2]     = mu;
        gn[bg * 2 + 1] = rsqrtf(var + 1e-5f);
    }
}

__global__ void k_gn_apply(float* __restrict__ y, const float* __restrict__ gn,
                           const float* __restrict__ gam, const float* __restrict__ bet,
                           int HW, int total) {
    int i = blockIdx.x * blockDim.x + threadIdx.x;
    if (i >= total) return;
    int c = (i / HW) & 255;
    int b = i / (256 * HW);
    int g = c >> 3;
    float mu = gn[(b * 32 + g) * 2], rs = gn[(b * 32 + g) * 2 + 1];
    y[i] = (y[i] - mu) * rs * gam[c] + bet[c];
}

// ---------------------------------------------------------------------------
extern "C" void kernel_launch(void* const* d_in, const int* in_sizes, int n_in,
                              void* d_out, int out_size, void* d_ws, size_t ws_size,
                              hipStream_t stream) {
    const float* feat_3 = (const float*)d_in[1];   // [4,256,64,64]
    const float* feat_4 = (const float*)d_in[2];   // [4,256,32,32]
    const float* feat_5 = (const float*)d_in[3];   // [4,256,16,16]
    const float* w_off5 = (const float*)d_in[4];
    const float* w_om5  = (const float*)d_in[5];
    const float* b_om5  = (const float*)d_in[6];
    const float* w_dcn5 = (const float*)d_in[7];
    const float* b_dcn5 = (const float*)d_in[8];
    const float* w_c1   = (const float*)d_in[9];
    const float* w_off4 = (const float*)d_in[10];
    const float* w_om4  = (const float*)d_in[11];
    const float* b_om4  = (const float*)d_in[12];
    const float* w_dcn4 = (const float*)d_in[13];
    const float* b_dcn4 = (const float*)d_in[14];
    const float* w_sa   = (const float*)d_in[15];
    const float* bn_g   = (const float*)d_in[16];
    const float* bn_b   = (const float*)d_in[17];
    const float* bn_m   = (const float*)d_in[18];
    const float* bn_v   = (const float*)d_in[19];
    const float* w_sem  = (const float*)d_in[20];
    const float* gn_g   = (const float*)d_in[21];
    const float* gn_b   = (const float*)d_in[22];
    float* out = (float*)d_out;                    // [4,256,128,128]
    float* ws  = (float*)d_ws;

    // workspace layout (floats). Stage-4/5 temporaries are dead before FEAT2
    // is written, so they alias the FEAT2 region (33,554,432 floats at 0).
    float* UP5   = ws + 0;           // [4,256,32,32]   1,048,576
    float* OFF5  = ws + 1048576;     // [4,256,32,32]   1,048,576
    float* OM5   = ws + 2097152;     // [4,216,32,32]     884,736
    float* F5A   = ws + 2981888;     // [4,256,32,32]   1,048,576
    float* F4N   = ws + 4030464;     // [4,256,32,32]   1,048,576
    float* UP4   = ws + 5079040;     // [4,256,64,64]   4,194,304
    float* OFF4  = ws + 9273344;     // [4,256,64,64]   4,194,304
    float* OM4   = ws + 13467648;    // [4,216,64,64]   3,538,944
    float* FEAT2 = ws + 0;           // [4,512,128,128] 33,554,432 (aliases above)
    float* F4A   = ws + 33554432;    // [4,256,64,64]   4,194,304
    float* AVG   = ws + 37748736;    // [4,128,128]        65,536
    float* MAX   = ws + 37814272;    // [4,128,128]        65,536
    float* SA    = ws + 37879808;    // [4,128,128]        65,536
    float* GN    = ws + 37945344;    // [4,32,2]              256

    const int B = 4;
    // ---- stage 5 -> 4 (32x32, HW=1024) ----
    k_bilinear<<<(B*256*32*32 + 255) / 256, 256, 0, stream>>>(
        feat_5, UP5, B, 256, 16, 16, 32, 32, 256*256, 256*1024);
    k_conv1x1_wmma<1024, false><<<(B*16*32) / 8, 256, 0, stream>>>(
        w_off5, feat_4, UP5, 256*1024, 256*1024, 2.0f, nullptr,
        OFF5, 256*1024, B);
    k_conv3x3_wmma<32, 32><<<(B*14*32) / 8, 256, 0, stream>>>(
        w_om5, b_om5, OFF5, OM5, B);
    k_dcn_wmma<32, 32><<<B*2*32, 256, 0, stream>>>(
        UP5, OM5, w_dcn5, b_dcn5, F5A, B);
    k_conv1x1_wmma<1024, false><<<(B*16*32) / 8, 256, 0, stream>>>(
        w_c1, F5A, feat_4, 256*1024, 256*1024, 1.0f, nullptr,
        F4N, 256*1024, B);

    // ---- stage 4 -> 3 (64x64, HW=4096) ----
    k_bilinear<<<(B*256*64*64 + 255) / 256, 256, 0, stream>>>(
        F4N, UP4, B, 256, 32, 32, 64, 64, 256*1024, 256*4096);
    k_conv1x1_wmma<4096, false><<<(B*16*128) / 8, 256, 0, stream>>>(
        w_off4, feat_3, UP4, 256*4096, 256*4096, 2.0f, nullptr,
        OFF4, 256*4096, B);
    k_conv3x3_wmma<64, 64><<<(B*14*128) / 8, 256, 0, stream>>>(
        w_om4, b_om4, OFF4, OM4, B);
    k_dcn_wmma<64, 64><<<B*2*128, 256, 0, stream>>>(
        UP4, OM4, w_dcn4, b_dcn4, F4A, B);

    // ---- feat_2 = resize(concat(f4align, feat_3), 128x128) ----
    k_bilinear<<<(B*256*128*128 + 255) / 256, 256, 0, stream>>>(
        F4A, FEAT2, B, 256, 64, 64, 128, 128, 256*4096, 512*16384);
    k_bilinear<<<(B*256*128*128 + 255) / 256, 256, 0, stream>>>(
        feat_3, FEAT2 + 256*16384, B, 256, 64, 64, 128, 128, 256*4096, 512*16384);

    // ---- SEM: spatial attention + 1x1 conv + GroupNorm ----
    k_avgmax<<<(B*16384 + 255) / 256, 256, 0, stream>>>(FEAT2, AVG, MAX, B, 16384);
    k_sa<<<(B*16384 + 255) / 256, 256, 0, stream>>>(
        AVG, MAX, w_sa, bn_g, bn_b, bn_m, bn_v, SA, B, 128, 128);
    k_conv1x1_wmma<16384, true><<<(B*16*512) / 8, 256, 0, stream>>>(
        w_sem, FEAT2, FEAT2 + 256*16384, 512*16384, 512*16384, 1.0f, SA,
        out, 256*16384, B);
    k_gn_reduce<<<B*32, 256, 0, stream>>>(out, GN, 16384);
    k_gn_apply<<<(B*256*16384 + 255) / 256, 256, 0, stream>>>(
        out, GN, gn_g, gn_b, 16384, B*256*16384);
}